// PostProcessor_58205396795712
// MI455X (gfx1250) — compile-verified
//
#include <hip/hip_runtime.h>
#include <stdint.h>

// ---------------- problem constants ----------------
#define NN    16
#define CC    80
#define HH    200
#define WW    152
#define HW    (HH * WW)          // 30400
#define KTOP  100
#define NBINS 4096               // key>>18 for positive floats < 1.0 -> max 4064
#define CAP   1024               // per-image candidate buffer (power of two, bitonic)
#define TH_LOGIT (-1.3862944f)   // ln(0.2/0.8): sigmoid(x) > 0.2  <=>  x > this
#define NMS_TH   0.6f

// scan decomposition: per image CC*HW = 2,432,000 floats = 608,000 float4
#define BPI          95          // blocks per image
#define F4_PER_BLOCK 6400        // 608000 / 95
#define SCAN_ITERS   25          // 6400 / 256 threads

__device__ __forceinline__ float sigm(float x) {
  return 1.0f / (1.0f + __expf(-x));
}

// generic pointer to LDS: low 32 bits are the LDS byte address
__device__ __forceinline__ uint32_t lds_off(const void* p) {
  return (uint32_t)(uintptr_t)p;
}

// CDNA5 async copy: 16 bytes from global vaddr go to the LDS byte address in vdst, tracked by ASYNCcnt
__device__ __forceinline__ void async_ld_b128(uint32_t lds, const void* gp) {
  asm volatile("global_load_async_to_lds_b128 %0, %1, off"
               :: "v"(lds), "v"((uint64_t)(uintptr_t)gp)
               : "memory");
}
__device__ __forceinline__ void wait_async0() {
  asm volatile("s_wait_asynccnt 0x0" ::: "memory");
}
__device__ __forceinline__ void wait_async1() {
  asm volatile("s_wait_asynccnt 0x1" ::: "memory");
}

// ---------------- kernel 0: zero workspace ----------------
__global__ void zero_kernel(uint32_t* p, int nwords) {
  int i = blockIdx.x * blockDim.x + threadIdx.x;
  if (i < nwords) p[i] = 0u;
}

// ---------------- kernel 1/3: streaming scan ----------------
// phase 0: build per-image 4096-bin histogram of score-key
// phase 1: collect all candidates with bin >= threshBin[n] into cand[]
__global__ __launch_bounds__(256)
void scan_kernel(int phase,
                 const float* __restrict__ cls,
                 const float* __restrict__ ctr,
                 uint32_t* __restrict__ ghist,
                 const uint32_t* __restrict__ threshBin,
                 uint32_t* __restrict__ candCount,
                 uint64_t* __restrict__ cand)
{
  __shared__ uint32_t lhist[NBINS];
  __shared__ float4   stage[2][256];

  const int n = blockIdx.y;
  const int t = threadIdx.x;

  if (phase == 0) {
    for (int i = t; i < NBINS; i += 256) lhist[i] = 0u;
    __syncthreads();
  }
  uint32_t bmin = 0;
  if (phase == 1) bmin = threshBin[n];

  const float4* cls4 = (const float4*)(cls + (size_t)n * CC * HW);
  const float4* ctr4 = (const float4*)(ctr + (size_t)n * HW);
  const int g0 = blockIdx.x * F4_PER_BLOCK + t;

  const uint32_t lds0 = lds_off(&stage[0][t]);
  const uint32_t lds1 = lds_off(&stage[1][t]);

  // software pipeline: async-stage box_cls tiles through LDS (double buffer)
  async_ld_b128(lds0, cls4 + g0);

  for (int it = 0; it < SCAN_ITERS; ++it) {
    const int cur = it & 1;
    const int g   = g0 + it * 256;
    if (it + 1 < SCAN_ITERS) {
      async_ld_b128(cur ? lds0 : lds1, cls4 + g + 256);
      wait_async1();               // wait for tile `it`, keep `it+1` in flight
    } else {
      wait_async0();
    }
    float4 v = stage[cur][t];      // ds_load_b128 (own slot; no barrier needed)

    const int e0 = g * 4;          // element index in [CC*HW) (class-major)
    const int l0 = e0 % HW;        // HW % 4 == 0 -> one class per float4
    float4 cv = ctr4[l0 >> 2];

    const float lg[4] = { v.x, v.y, v.z, v.w };
    const float cn[4] = { cv.x, cv.y, cv.z, cv.w };
    #pragma unroll
    for (int j = 0; j < 4; ++j) {
      if (lg[j] > TH_LOGIT) {                       // sigmoid > 0.2 on raw logit
        float comb = sigm(lg[j]) * sigm(cn[j]);     // (0,1]: positive float
        uint32_t key = __float_as_uint(comb);       // monotone for positives
        uint32_t bin = key >> 18;
        if (phase == 0) {
          atomicAdd(&lhist[bin], 1u);
        } else if (bin >= bmin) {
          uint32_t pos = atomicAdd(&candCount[n], 1u);
          if (pos < CAP) {
            uint32_t idx = (uint32_t)(e0 + j);      // c*HW + l
            cand[(size_t)n * CAP + pos] =
                ((uint64_t)key << 32) | (uint64_t)(0xFFFFFFFFu - idx);
          }
        }
      }
    }
  }

  if (phase == 0) {
    __syncthreads();
    for (int i = t; i < NBINS; i += 256) {
      uint32_t cnt = lhist[i];
      if (cnt) atomicAdd(&ghist[(size_t)n * NBINS + i], cnt);
    }
  }
}

// ---------------- kernel 2: parallel per-image threshold-bin search ----------
// One block per image. Thread t owns a 16-bin chunk, chunks ordered top-down.
// Exclusive prefix over chunk sums gives "count strictly above chunk"; the one
// straddling thread (cum < K <= cum+chunkSum) localizes the K-th-score bin.
__global__ __launch_bounds__(256)
void thresh_kernel(const uint32_t* __restrict__ ghist,
                   uint32_t* __restrict__ threshBin)
{
  __shared__ uint32_t csum[256];
  const int n = blockIdx.x;
  const int t = threadIdx.x;
  const uint32_t* h = ghist + (size_t)n * NBINS;

  const int hi = NBINS - 16 * t;      // exclusive top of this thread's chunk
  uint32_t b[16];
  uint32_t s = 0;
  #pragma unroll
  for (int j = 0; j < 16; ++j) { b[j] = h[hi - 1 - j]; s += b[j]; }
  csum[t] = s;
  __syncthreads();

  // Hillis-Steele inclusive scan over chunk sums (chunk 0 = topmost bins)
  uint32_t v = s;
  for (int ofs = 1; ofs < 256; ofs <<= 1) {
    uint32_t w = (t >= ofs) ? csum[t - ofs] : 0u;
    __syncthreads();
    v += w;
    csum[t] = v;
    __syncthreads();
  }

  uint32_t cum = v - s;               // count of keys in all bins above chunk
  if (cum < KTOP && cum + s >= KTOP) {
    #pragma unroll
    for (int j = 0; j < 16; ++j) {
      uint32_t c = b[j];
      if (cum + c >= KTOP) { threshBin[n] = (uint32_t)(hi - 1 - j); break; }
      cum += c;
    }
  }
  // total < K: no thread writes; threshBin stays 0 -> collect all candidates
}

// ---------------- kernel 4: sort, decode, NMS, write ----------------
__global__ __launch_bounds__(256)
void final_kernel(const float* __restrict__ loc,
                  const float* __restrict__ reg,
                  const uint32_t* __restrict__ candCount,
                  const uint64_t* __restrict__ cand,
                  const int* __restrict__ size_h_p,
                  const int* __restrict__ size_w_p,
                  float* __restrict__ out)
{
  __shared__ uint64_t      sc[CAP];
  __shared__ float         sbox[KTOP][4];   // unoffset (output) boxes
  __shared__ float         sbo[KTOP][4];    // class-offset boxes for IoU
  __shared__ float         sarea[KTOP];
  __shared__ float         sscore[KTOP];
  __shared__ int           slab[KTOP];
  __shared__ unsigned char svalid[KTOP];
  __shared__ unsigned char skeep[KTOP];
  __shared__ int           ssup;

  const int n = blockIdx.x;
  const int t = threadIdx.x;

  uint32_t M = candCount[n];
  if (M > CAP) M = CAP;

  for (int i = t; i < CAP; i += 256)
    sc[i] = (i < (int)M) ? cand[(size_t)n * CAP + i] : 0ull;
  __syncthreads();

  // bitonic sort, descending: key desc, then index asc (via ~idx in low bits)
  for (int k = 2; k <= CAP; k <<= 1) {
    for (int j = k >> 1; j > 0; j >>= 1) {
      for (int i = t; i < CAP; i += 256) {
        int p = i ^ j;
        if (p > i) {
          uint64_t a = sc[i], b = sc[p];
          bool desc = ((i & k) == 0);
          if (desc ? (a < b) : (a > b)) { sc[i] = b; sc[p] = a; }
        }
      }
      __syncthreads();
    }
  }

  const int   sh = size_h_p[0], sw = size_w_p[0];
  const float wmax = (float)(sw - 1);
  const float hmax = (float)(sh - 1);
  const float offscale = (float)(sh > sw ? sh : sw);

  if (t < KTOP) {
    uint64_t v   = sc[t];
    uint32_t key = (uint32_t)(v >> 32);
    bool valid   = (t < (int)M) && (key != 0u);
    uint32_t idx = valid ? (0xFFFFFFFFu - (uint32_t)(v & 0xFFFFFFFFu)) : 0u;
    int c = (int)(idx / HW);
    int l = (int)(idx % HW);

    float px = loc[2 * l + 0], py = loc[2 * l + 1];
    const float* rb = reg + (size_t)n * 4 * HW;
    float rl = rb[0 * HW + l], rt = rb[1 * HW + l];
    float rr = rb[2 * HW + l], rB = rb[3 * HW + l];

    float x1 = fminf(fmaxf(px - rl, 0.f), wmax);
    float y1 = fminf(fmaxf(py - rt, 0.f), hmax);
    float x2 = fminf(fmaxf(px + rr, 0.f), wmax);
    float y2 = fminf(fmaxf(py + rB, 0.f), hmax);
    valid = valid && (x2 - x1 >= 0.f) && (y2 - y1 >= 0.f);

    float comb = __uint_as_float(key);
    int   lab  = c + 1;
    float off  = (float)lab * offscale;

    sbox[t][0] = x1; sbox[t][1] = y1; sbox[t][2] = x2; sbox[t][3] = y2;
    sbo[t][0] = x1 + off; sbo[t][1] = y1 + off;
    sbo[t][2] = x2 + off; sbo[t][3] = y2 + off;
    sarea[t]  = fmaxf(x2 - x1, 0.f) * fmaxf(y2 - y1, 0.f);
    sscore[t] = valid ? sqrtf(comb) : 0.f;
    slab[t]   = lab;
    svalid[t] = valid ? 1 : 0;
    skeep[t]  = 0;
  }
  __syncthreads();

  // sequential greedy NMS (i ordered by score), suppression test parallel over j<i
  for (int i = 0; i < KTOP; ++i) {
    if (t == 0) ssup = 0;
    __syncthreads();
    if (t < i && skeep[t]) {
      float xx1 = fmaxf(sbo[i][0], sbo[t][0]);
      float yy1 = fmaxf(sbo[i][1], sbo[t][1]);
      float xx2 = fminf(sbo[i][2], sbo[t][2]);
      float yy2 = fminf(sbo[i][3], sbo[t][3]);
      float iw = fmaxf(xx2 - xx1, 0.f), ih = fmaxf(yy2 - yy1, 0.f);
      float inter = iw * ih;
      float iou = inter / (sarea[i] + sarea[t] - inter + 1e-9f);
      if (iou > NMS_TH) atomicOr(&ssup, 1);
    }
    __syncthreads();
    if (t == 0) skeep[i] = (svalid[i] && ssup == 0) ? 1 : 0;
    __syncthreads();
  }

  // outputs concatenated flat in reference return order (all as float)
  if (t < KTOP) {
    float* ob = out;                       // boxes  [NN][KTOP][4]
    float* os = out + NN * KTOP * 4;       // scores [NN][KTOP]
    float* ol = os + NN * KTOP;            // labels [NN][KTOP]
    float* ok = ol + NN * KTOP;            // keep   [NN][KTOP]
    int o = n * KTOP + t;
    ob[o * 4 + 0] = sbox[t][0];
    ob[o * 4 + 1] = sbox[t][1];
    ob[o * 4 + 2] = sbox[t][2];
    ob[o * 4 + 3] = sbox[t][3];
    os[o] = skeep[t] ? sscore[t] : 0.f;
    ol[o] = (float)slab[t];
    ok[o] = skeep[t] ? 1.f : 0.f;
  }
}

// ---------------- host launcher ----------------
extern "C" void kernel_launch(void* const* d_in, const int* in_sizes, int n_in,
                              void* d_out, int out_size, void* d_ws, size_t ws_size,
                              hipStream_t stream) {
  const float* loc = (const float*)d_in[0];   // [HW,2]
  const float* cls = (const float*)d_in[1];   // [N,C,H,W]
  const float* reg = (const float*)d_in[2];   // [N,4,H,W]
  const float* ctr = (const float*)d_in[3];   // [N,1,H,W]
  const int*   shp = (const int*)d_in[4];     // scalar size_h
  const int*   swp = (const int*)d_in[5];     // scalar size_w
  float* out = (float*)d_out;

  // workspace layout: [hist NN*NBINS u32][candCount NN u32][threshBin NN u32][cand NN*CAP u64]
  uint32_t* ghist     = (uint32_t*)d_ws;
  uint32_t* candCount = ghist + (size_t)NN * NBINS;
  uint32_t* threshBin = candCount + NN;
  uint64_t* cand      = (uint64_t*)(threshBin + NN);   // byte offset divisible by 8

  const int zwords = NN * NBINS + 2 * NN;
  zero_kernel<<<(zwords + 255) / 256, 256, 0, stream>>>((uint32_t*)d_ws, zwords);

  dim3 gscan(BPI, NN);
  scan_kernel<<<gscan, 256, 0, stream>>>(0, cls, ctr, ghist, threshBin, candCount, cand);
  thresh_kernel<<<NN, 256, 0, stream>>>(ghist, threshBin);
  scan_kernel<<<gscan, 256, 0, stream>>>(1, cls, ctr, ghist, threshBin, candCount, cand);
  final_kernel<<<NN, 256, 0, stream>>>(loc, reg, candCount, cand, shp, swp, out);
}